// BioInspiredESN_62697932587184
// MI455X (gfx1250) — compile-verified
//
#include <hip/hip_runtime.h>
#include <math.h>

// ---------------- problem constants (from reference) ----------------
#define R_DIM   8192
#define CORE_N  4915        // int(8192*0.6)
#define SHELL_N 2457        // int(8192*0.3)
#define FB_N    820         // R - CORE - SHELL
#define H_DIM   1024
#define G_DIM   (4 * H_DIM) // 4096 LSTM gate rows
#define LEAK    0.1f

#define NT_CORE  ((CORE_N  + 15) / 16)   // 308 row-tiles
#define NT_SHELL ((SHELL_N + 15) / 16)   // 154
#define NT_FB    ((FB_N    + 15) / 16)   // 52
#define NT_G     (G_DIM / 16)            // 256 (exact)

// split-K factors per GEMV (chunks ~1K-1.2K, boundaries %32==0)
#define NC_CORE   4
#define NC_SHELL  2
#define NC_FB     1
#define NC_IH     8
#define NC_HH     1

// stage wave counts
#define ST1_WAVES (NT_CORE * NC_CORE + NT_SHELL * NC_SHELL + NT_FB * NC_FB) // 1592
#define ST2_WAVES (NT_CORE * NC_SHELL + NT_CORE * NC_FB)                    // 924
#define ST3_WAVES (NT_G * NC_IH + NT_G * NC_HH)                             // 2304

typedef __attribute__((ext_vector_type(2))) float v2f;
typedef __attribute__((ext_vector_type(8))) float v8f;

__device__ __forceinline__ float sigmoidf_(float v) { return 1.0f / (1.0f + expf(-v)); }

__device__ __forceinline__ void chunk_bounds(int K, int NC, int chunk,
                                             int& kbeg, int& kend)
{
    const int base = ((K / NC) / 32) * 32;   // %32 so only last chunk has a tail
    kbeg = chunk * base;
    kend = (chunk == NC - 1) ? K : kbeg + base;
}

// ---------------------------------------------------------------------------
// Wave-level partial GEMV via V_WMMA_F32_16X16X4_F32 (wave32, CDNA5).
// y[m] = dot(W[row0+m, kbeg:kend], x[kbeg:kend]) for m = 0..15 of row-major W.
// A operand = 16x4 fp32 W tile in the ISA A-layout (lane L<16: M=L, K pair
// {0,1}; lane L>=16: M=L-16, K pair {2,3}) -- each lane loads exactly its two
// dwords (merges to one b64 with immediate offsets off an incrementing ptr).
// B trick: we only read D column 0, and D[:,j] depends only on B[:,j] -- so
// every lane loads the SAME x pair for its kh (B[k][n] = x[k0+k] for all n).
// No column masking, no selects, no divergence; lanes 0-15 hit identical
// addresses (single fetch + broadcast). All D columns then hold the result;
// we extract column 0: lane 0 -> M 0..7 in c[0..7], lane 16 -> M 8..15.
// Loop bounds are readfirstlane'd => scalar loop control, EXEC all-1s at
// every WMMA. Tail: zero-fill A only (0 * finite x = 0), clamped B loads.
// ---------------------------------------------------------------------------
__device__ __forceinline__ v8f wmma_gemv_part(const float* __restrict__ W, int ldw,
                                              int row0, int nrows, int K,
                                              int kbeg, int kend,
                                              const float* __restrict__ x,
                                              unsigned lane)
{
    const int rit = (int)(lane & 15u);   // row within tile
    const int kh  = (int)(lane >> 4);    // 0 -> K pair {0,1}, 1 -> {2,3}
    const int r   = rit < nrows ? rit : (nrows - 1);
    const float* __restrict__ Arow = W + (size_t)(row0 + r) * (size_t)ldw;

    v8f c = {0.f, 0.f, 0.f, 0.f, 0.f, 0.f, 0.f, 0.f};

    int       k0 = __builtin_amdgcn_readfirstlane(kbeg);   // wave-uniform
    const int ke = __builtin_amdgcn_readfirstlane(kend);   // -> scalar loop
    const int km = k0 + (((ke - k0) >> 5) << 5);           // unroll-8 main

    const float* __restrict__ pA = Arow + k0 + 2 * kh;
    const float* __restrict__ px = x    + k0 + 2 * kh;

    for (; k0 < km; k0 += 32, pA += 32, px += 32) {
#pragma unroll
        for (int u = 0; u < 8; ++u) {
            v2f a, b;
            a.x = pA[4 * u];          // base + imm offset, merges to b64
            a.y = pA[4 * u + 1];
            b.x = px[4 * u];          // same for all lanes of a kh half
            b.y = px[4 * u + 1];
            c = __builtin_amdgcn_wmma_f32_16x16x4_f32(false, a, false, b,
                                                      (short)0, c, false, false);
        }
    }
    for (; k0 < ke; k0 += 4) {        // tail (only last chunk; K % 4 too)
        const int kb = k0 + 2 * kh;
        const int i0 = (kb     < K) ? kb     : (K - 1);  // clamped, in-bounds
        const int i1 = (kb + 1 < K) ? kb + 1 : (K - 1);
        const float a0 = Arow[i0], a1 = Arow[i1];
        v2f a, b;
        a.x = (kb     < K) ? a0 : 0.0f;   // zero-fill A beyond K
        a.y = (kb + 1 < K) ? a1 : 0.0f;
        b.x = x[i0];                      // clamped, finite: 0*x stays 0
        b.y = x[i1];
        c = __builtin_amdgcn_wmma_f32_16x16x4_f32(false, a, false, b,
                                                  (short)0, c, false, false);
    }
    return c;
}

// extract D column 0 (lanes 0 and 16) into parts[(rbase+m)*stride + slot]
__device__ __forceinline__ void store_col0(v8f c, unsigned lane, int nrows,
                                           float* __restrict__ parts,
                                           int rbase, int stride, int slot)
{
    if ((lane & 15u) == 0u) {
        const int mb = (int)(lane >> 4) * 8;
        for (int v = 0; v < 8; ++v) {
            const int m = mb + v;
            if (m < nrows)
                parts[(size_t)(rbase + m) * stride + slot] = c[v];
        }
    }
}

// ---------------------------------------------------------------------------
// Stage 1: split-K partials of core@Wc^T, shell@Ws^T, fb@Wf^T.
// partsA[g*4 + chunk], g = global reservoir row.
// ---------------------------------------------------------------------------
__global__ void k_preds_parts(const float* __restrict__ state,
                              const float* __restrict__ W_core,
                              const float* __restrict__ W_shell,
                              const float* __restrict__ W_feedback,
                              float* __restrict__ partsA)
{
    const unsigned wave = blockIdx.x * (blockDim.x >> 5) + (threadIdx.x >> 5);
    const unsigned lane = threadIdx.x & 31u;
    if (wave >= ST1_WAVES) return;

    const float* W; const float* xv;
    int K, NC, tile, chunk, sect_base, sectN;
    if (wave < NT_CORE * NC_CORE) {
        W = W_core; xv = state; K = CORE_N; NC = NC_CORE;
        sect_base = 0; sectN = CORE_N;
        tile = (int)wave / NC_CORE; chunk = (int)wave % NC_CORE;
    } else if (wave < NT_CORE * NC_CORE + NT_SHELL * NC_SHELL) {
        const int w2 = (int)wave - NT_CORE * NC_CORE;
        W = W_shell; xv = state + CORE_N; K = SHELL_N; NC = NC_SHELL;
        sect_base = CORE_N; sectN = SHELL_N;
        tile = w2 / NC_SHELL; chunk = w2 % NC_SHELL;
    } else {
        const int w2 = (int)wave - (NT_CORE * NC_CORE + NT_SHELL * NC_SHELL);
        W = W_feedback; xv = state + CORE_N + SHELL_N; K = FB_N; NC = NC_FB;
        sect_base = CORE_N + SHELL_N; sectN = FB_N;
        tile = w2; chunk = 0;
    }
    const int row0  = tile * 16;
    const int nrows = (sectN - row0 < 16) ? (sectN - row0) : 16;

    int kbeg, kend;
    chunk_bounds(K, NC, chunk, kbeg, kend);
    v8f c = wmma_gemv_part(W, K, row0, nrows, K, kbeg, kend, xv, lane);
    store_col0(c, lane, nrows, partsA, sect_base + row0, 4, chunk);
}

// Stage 1 reduce: pred[g] = tanh(sum(parts) [+ W_in*x for core rows]).
__global__ void k_preds_reduce(const float* __restrict__ partsA,
                               const float* __restrict__ W_in,
                               const float* __restrict__ x,
                               float* __restrict__ pred)
{
    const int g = blockIdx.x * blockDim.x + threadIdx.x;
    if (g >= R_DIM) return;
    const int nc = (g < CORE_N) ? NC_CORE : (g < CORE_N + SHELL_N) ? NC_SHELL : NC_FB;
    float acc = 0.0f;
    for (int cidx = 0; cidx < nc; ++cidx) acc += partsA[(size_t)g * 4 + cidx];
    if (g < CORE_N) acc += W_in[g] * x[0];
    pred[g] = tanhf(acc);
}

// ---------------------------------------------------------------------------
// Stage 2: partials of shell_pred@Wsc^T (slots 0..1) and fb_pred@Wfc^T (slot 2)
// for the CORE rows. partsB[i*3 + slot].
// ---------------------------------------------------------------------------
__global__ void k_core_parts(const float* __restrict__ pred,
                             const float* __restrict__ W_shell_core,
                             const float* __restrict__ W_feedback_core,
                             float* __restrict__ partsB)
{
    const unsigned wave = blockIdx.x * (blockDim.x >> 5) + (threadIdx.x >> 5);
    const unsigned lane = threadIdx.x & 31u;
    if (wave >= ST2_WAVES) return;

    const float* W; const float* xv;
    int K, NC, tile, chunk, slot;
    if (wave < NT_CORE * NC_SHELL) {
        W = W_shell_core; xv = pred + CORE_N; K = SHELL_N; NC = NC_SHELL;
        tile = (int)wave / NC_SHELL; chunk = (int)wave % NC_SHELL; slot = chunk;
    } else {
        const int w2 = (int)wave - NT_CORE * NC_SHELL;
        W = W_feedback_core; xv = pred + CORE_N + SHELL_N; K = FB_N; NC = NC_FB;
        tile = w2; chunk = 0; slot = 2;
    }
    const int row0  = tile * 16;
    const int nrows = (CORE_N - row0 < 16) ? (CORE_N - row0) : 16;

    int kbeg, kend;
    chunk_bounds(K, NC, chunk, kbeg, kend);
    v8f c = wmma_gemv_part(W, K, row0, nrows, K, kbeg, kend, xv, lane);
    store_col0(c, lane, nrows, partsB, row0, 3, slot);
}

// Stage 2 combine + leak + s-transform for core rows.
__global__ void k_core_combine(const float* __restrict__ state,
                               const float* __restrict__ pred,
                               const float* __restrict__ partsB,
                               const float* __restrict__ omega,
                               const float* __restrict__ gam,
                               const float* __restrict__ dopamine,
                               float* __restrict__ s_out)
{
    const int i = blockIdx.x * blockDim.x + threadIdx.x;
    if (i >= CORE_N) return;
    const float t1 = tanhf(partsB[(size_t)i * 3 + 0] + partsB[(size_t)i * 3 + 1]);
    const float t2 = tanhf(partsB[(size_t)i * 3 + 2]);
    const float cp = pred[i] + t1 + t2;
    const float cn = (1.0f - LEAK) * state[i] + LEAK * cp;
    const float sv = cn + 0.1f * (omega[i] * cn - gam[i] * tanhf(cn));
    s_out[i] = dopamine[0] * sv;
}

// shell + fb rows: leak + s-transform only (pred already tanh'ed).
__global__ void k_shellfb_update(const float* __restrict__ state,
                                 const float* __restrict__ pred,
                                 const float* __restrict__ omega,
                                 const float* __restrict__ gam,
                                 const float* __restrict__ dopamine,
                                 float* __restrict__ s_out)
{
    const int i = CORE_N + blockIdx.x * blockDim.x + threadIdx.x;
    if (i >= R_DIM) return;
    const float cn = (1.0f - LEAK) * state[i] + LEAK * pred[i];
    const float sv = cn + 0.1f * (omega[i] * cn - gam[i] * tanhf(cn));
    s_out[i] = dopamine[0] * sv;
}

// ---------------------------------------------------------------------------
// Stage 3: partials of s@W_ih^T (slots 0..7) and h@W_hh^T (slot 8).
// partsC[i*9 + slot], i = gate row.
// ---------------------------------------------------------------------------
__global__ void k_gates_parts(const float* __restrict__ s,
                              const float* __restrict__ h,
                              const float* __restrict__ W_ih,
                              const float* __restrict__ W_hh,
                              float* __restrict__ partsC)
{
    const unsigned wave = blockIdx.x * (blockDim.x >> 5) + (threadIdx.x >> 5);
    const unsigned lane = threadIdx.x & 31u;
    if (wave >= ST3_WAVES) return;

    const float* W; const float* xv;
    int K, NC, tile, chunk, slot;
    if (wave < NT_G * NC_IH) {
        W = W_ih; xv = s; K = R_DIM; NC = NC_IH;
        tile = (int)wave / NC_IH; chunk = (int)wave % NC_IH; slot = chunk;
    } else {
        const int w2 = (int)wave - NT_G * NC_IH;
        W = W_hh; xv = h; K = H_DIM; NC = NC_HH;
        tile = w2; chunk = 0; slot = 8;
    }
    const int row0 = tile * 16;   // G_DIM % 16 == 0 -> always full tiles

    int kbeg, kend;
    chunk_bounds(K, NC, chunk, kbeg, kend);
    v8f c = wmma_gemv_part(W, K, row0, 16, K, kbeg, kend, xv, lane);
    store_col0(c, lane, 16, partsC, row0, 9, slot);
}

// Stage 3 reduce: gates[i] = sum(parts) + b_ih + b_hh.
__global__ void k_gates_reduce(const float* __restrict__ partsC,
                               const float* __restrict__ b_ih,
                               const float* __restrict__ b_hh,
                               float* __restrict__ gates)
{
    const int i = blockIdx.x * blockDim.x + threadIdx.x;
    if (i >= G_DIM) return;
    float acc = b_ih[i] + b_hh[i];
    for (int cidx = 0; cidx < 9; ++cidx) acc += partsC[(size_t)i * 9 + cidx];
    gates[i] = acc;
}

// LSTM cell elementwise.
__global__ void k_lstm(const float* __restrict__ gates,
                       const float* __restrict__ c_in,
                       float* __restrict__ h_out,
                       float* __restrict__ c_out)
{
    const int k = blockIdx.x * blockDim.x + threadIdx.x;
    if (k >= H_DIM) return;
    const float ig = sigmoidf_(gates[k]);
    const float fg = sigmoidf_(gates[H_DIM + k]);
    const float gg = tanhf(gates[2 * H_DIM + k]);
    const float og = sigmoidf_(gates[3 * H_DIM + k]);
    const float cn = fg * c_in[k] + ig * gg;
    c_out[k] = cn;
    h_out[k] = og * tanhf(cn);
}

// output = h_new @ W_out^T + b_out (single 1024-dot reduction).
__global__ void k_output(const float* __restrict__ h_new,
                         const float* __restrict__ W_out,
                         const float* __restrict__ b_out,
                         float* __restrict__ out)
{
    __shared__ float red[256];
    float acc = 0.0f;
    for (int k = threadIdx.x; k < H_DIM; k += 256)
        acc += h_new[k] * W_out[k];
    red[threadIdx.x] = acc;
    __syncthreads();
    for (int off = 128; off > 0; off >>= 1) {
        if ((int)threadIdx.x < off) red[threadIdx.x] += red[threadIdx.x + off];
        __syncthreads();
    }
    if (threadIdx.x == 0) out[0] = red[0] + b_out[0];
}

// ---------------------------------------------------------------------------
// d_out layout: [0] output, [1:8193] s, [8193:9217] h_new, [9217:10241] c_new.
// ws layout (floats): pred 8192 | parts 36864 (reused per stage) | gates 4096.
// ---------------------------------------------------------------------------
extern "C" void kernel_launch(void* const* d_in, const int* in_sizes, int n_in,
                              void* d_out, int out_size, void* d_ws, size_t ws_size,
                              hipStream_t stream)
{
    (void)in_sizes; (void)n_in; (void)out_size; (void)ws_size;
    const float* x          = (const float*)d_in[0];
    const float* state      = (const float*)d_in[1];
    const float* lstm_h     = (const float*)d_in[2];
    const float* lstm_c     = (const float*)d_in[3];
    const float* W_in       = (const float*)d_in[4];
    const float* W_core     = (const float*)d_in[5];
    const float* W_shell    = (const float*)d_in[6];
    const float* W_feedback = (const float*)d_in[7];
    const float* W_sc       = (const float*)d_in[8];
    const float* W_fc       = (const float*)d_in[9];
    const float* dopamine   = (const float*)d_in[10];
    const float* omega      = (const float*)d_in[11];
    const float* gam        = (const float*)d_in[12];
    const float* W_ih       = (const float*)d_in[13];
    const float* W_hh       = (const float*)d_in[14];
    const float* b_ih       = (const float*)d_in[15];
    const float* b_hh       = (const float*)d_in[16];
    const float* W_out      = (const float*)d_in[17];
    const float* b_out      = (const float*)d_in[18];

    float* out   = (float*)d_out;
    float* s_out = out + 1;                 // 8192
    float* h_out = out + 1 + R_DIM;         // 1024
    float* c_out = out + 1 + R_DIM + H_DIM; // 1024

    float* ws    = (float*)d_ws;
    float* pred  = ws;                      // 8192
    float* parts = ws + R_DIM;              // 36864 (max of 32768/14745/36864)
    float* gates = ws + R_DIM + 36864;      // 4096

    const int WPB = 8;  // waves per 256-thread block (wave32)

    k_preds_parts<<<(ST1_WAVES + WPB - 1) / WPB, 256, 0, stream>>>(
        state, W_core, W_shell, W_feedback, parts);
    k_preds_reduce<<<(R_DIM + 255) / 256, 256, 0, stream>>>(parts, W_in, x, pred);

    k_core_parts<<<(ST2_WAVES + WPB - 1) / WPB, 256, 0, stream>>>(
        pred, W_sc, W_fc, parts);
    k_core_combine<<<(CORE_N + 255) / 256, 256, 0, stream>>>(
        state, pred, parts, omega, gam, dopamine, s_out);
    k_shellfb_update<<<((SHELL_N + FB_N) + 255) / 256, 256, 0, stream>>>(
        state, pred, omega, gam, dopamine, s_out);

    k_gates_parts<<<(ST3_WAVES + WPB - 1) / WPB, 256, 0, stream>>>(
        s_out, lstm_h, W_ih, W_hh, parts);
    k_gates_reduce<<<(G_DIM + 255) / 256, 256, 0, stream>>>(parts, b_ih, b_hh, gates);

    k_lstm<<<(H_DIM + 255) / 256, 256, 0, stream>>>(gates, lstm_c, h_out, c_out);
    k_output<<<1, 256, 0, stream>>>(h_out, W_out, b_out, out);
}